// PSM_10986526343601
// MI455X (gfx1250) — compile-verified
//
#include <hip/hip_runtime.h>

typedef __attribute__((ext_vector_type(2))) float v2f;
typedef __attribute__((ext_vector_type(8))) float v8f;

#define B_  2
#define C_  32
#define N_  1024   // H*W = 32*32

// Block: 256 threads = 8 wave32 waves. Each wave computes one 16-row i-tile.
// Grid: B*C*8 blocks; each block covers one (b,c) and 8 consecutive tiles.
__global__ __launch_bounds__(256)
void psm_attn_kernel(const float* __restrict__ feature,
                     const float* __restrict__ seg,
                     const float* __restrict__ gamma,
                     float* __restrict__ out)
{
    __shared__ __align__(16) float ldsM[N_];   // m[b, :]   (seg_map channel 1)
    __shared__ __align__(16) float ldsF[N_];   // f[b, c, :]

    const int bc   = blockIdx.x >> 3;      // (b*C + c)
    const int tg   = blockIdx.x & 7;       // tile group within (b,c)
    const int b    = bc >> 5;              // C_ == 32
    const int tid  = threadIdx.x;
    const int lane = tid & 31;
    const int wave = tid >> 5;

    const float* fptr = feature + (size_t)bc * N_;
    const float* mptr = seg + ((size_t)b * 2 + 1) * N_;   // channel 1

    for (int j = tid; j < N_; j += 256) {
        ldsM[j] = mptr[j];
        ldsF[j] = fptr[j];
    }
    __syncthreads();

    // Per-wave min/max of m (exact row-max of the rank-1 logits is s*mmax or s*mmin).
    float mx = -3.0e38f, mn = 3.0e38f;
    for (int j = lane; j < N_; j += 32) {
        const float v = ldsM[j];
        mx = fmaxf(mx, v);
        mn = fminf(mn, v);
    }
#pragma unroll
    for (int off = 16; off > 0; off >>= 1) {
        mx = fmaxf(mx, __shfl_xor(mx, off, 32));
        mn = fminf(mn, __shfl_xor(mn, off, 32));
    }

    const int half = lane >> 4;            // A-matrix K-half (K = 2*half + comp)
    const int row  = lane & 15;            // A-matrix M = row (both halves)
    const int i0   = (tg * 8 + wave) * 16; // first output row of this tile

    const float sM = ldsF[i0 + row];
    const float L2E  = 1.44269504088896340736f;
    const float sL2  = sM * L2E;
    const float rL2  = (sM >= 0.0f ? sM * mx : sM * mn) * L2E; // rowmax in log2 units
    const float g    = gamma[0];

    v8f acc_num = {};   // D[M,n] = sum_j f_j * exp(s_M m_j - rmax_M)  (all n equal)
    v8f acc_den = {};   // D[M,n] = sum_j       exp(s_M m_j - rmax_M)
    const v2f bOnes = {1.0f, 1.0f};   // all-ones B => D = row-sum of A, layout-proof

    const v2f* m2 = (const v2f*)ldsM;
    const v2f* f2 = (const v2f*)ldsF;

    for (int j0 = 0; j0 < N_; j0 += 4) {
        const int k = (j0 >> 1) + half;        // float2 index: elements j0+2*half, +1
        const v2f mv = m2[k];
        const v2f fv = f2[k];
        v2f e, ef;
        // Argument is in [-~10, 0] for this problem: safely above v_exp_f32's
        // -126 denorm guard, so use the raw instruction (no cndmask/ldexp fixup).
        e.x = __builtin_amdgcn_exp2f(sL2 * mv.x - rL2);
        e.y = __builtin_amdgcn_exp2f(sL2 * mv.y - rL2);
        ef.x = e.x * fv.x;
        ef.y = e.y * fv.y;
        // D = A x B + C   (16x16x4 f32 WMMA, wave32)
        acc_num = __builtin_amdgcn_wmma_f32_16x16x4_f32(
            false, ef, false, bOnes, (short)0, acc_num, false, false);
        acc_den = __builtin_amdgcn_wmma_f32_16x16x4_f32(
            false, e,  false, bOnes, (short)0, acc_den, false, false);
    }

    // C/D layout: acc[r] holds row M = r (lanes 0-15) or M = r+8 (lanes 16-31);
    // every column n carries the same row-sum, so each half already has its rows.
    float* obase = out + (size_t)bc * N_;
#pragma unroll
    for (int r = 0; r < 8; ++r) {
        if (row == r) {
            const int i = i0 + r + 8 * half;
            obase[i] = g * (acc_num[r] / acc_den[r]) + ldsF[i];
        }
    }
}

extern "C" void kernel_launch(void* const* d_in, const int* in_sizes, int n_in,
                              void* d_out, int out_size, void* d_ws, size_t ws_size,
                              hipStream_t stream) {
    (void)in_sizes; (void)n_in; (void)out_size; (void)d_ws; (void)ws_size;
    const float* feature = (const float*)d_in[0];
    const float* seg     = (const float*)d_in[1];
    const float* gamma   = (const float*)d_in[2];
    float* out           = (float*)d_out;

    dim3 grid(B_ * C_ * 8);
    dim3 block(256);
    psm_attn_kernel<<<grid, block, 0, stream>>>(feature, seg, gamma, out);
}